// LabelLayer_71837622993135
// MI455X (gfx1250) — compile-verified
//
#include <hip/hip_runtime.h>
#include <hip/hip_bf16.h>

// ---------------------------------------------------------------------------
// Problem: out[(img*512+p), :] = multihot(preds[img,:]) @ W.T + b
//   B=128 images, P=512 pairs, C=150 classes (padded to 152), D=1024 dims.
// All P rows of an image are identical -> compute Y[128,1024] once with
// fp32 WMMA (V_WMMA_F32_16X16X4_F32), then broadcast-store 256 MB with NT
// float4 stores (pure HBM-store-bound: ~11 us at 23.3 TB/s).
// ---------------------------------------------------------------------------

typedef __attribute__((ext_vector_type(2))) float v2f;
typedef __attribute__((ext_vector_type(4))) float v4f;
typedef __attribute__((ext_vector_type(8))) float v8f;

#define NUM_IMG 128
#define PAIRS   512
#define C_DIM   150
#define C_PAD   152   // pad K to multiple of 4 for 16x16x4 WMMA
#define D_DIM   1024

// ---------------------------------------------------------------------------
// Kernel 1: per-image multi-hot [128 x 152] fp32 into workspace.
// One block per image, 512 threads (one per pair). Racing 1.0 stores to the
// same LDS slot are idempotent -> no atomics needed.
// ---------------------------------------------------------------------------
__global__ void __launch_bounds__(PAIRS)
build_multihot(const int* __restrict__ preds, float* __restrict__ mh_out) {
    __shared__ float mh[C_PAD];
    const int img = blockIdx.x;
    const int t   = threadIdx.x;
    if (t < C_PAD) mh[t] = 0.0f;
    __syncthreads();
    const int p = preds[img * PAIRS + t];   // labels in 1..150
    mh[p - 1] = 1.0f;                       // benign same-value race
    __syncthreads();
    if (t < C_PAD) mh_out[img * C_PAD + t] = mh[t];
}

// ---------------------------------------------------------------------------
// Kernel 2: Y[128,1024] = MH[128,152] @ Wt[152,1024] + b via fp32 WMMA.
// 512 tiles of 16x16 (8 M-tiles x 64 N-tiles), one wave32 per tile,
// 38 x V_WMMA_F32_16X16X4_F32 per wave. Result row Y[img] is written
// directly to the p==0 output row (out row img*512), which must equal it.
//
// f32 16x16x4 VGPR layouts (ISA 7.12.2):
//   A 16x4 : m = lane&15, lane-half selects K pair {0,1}/{2,3}; v.x=K even, v.y=K odd
//   B 4x16 : n = lane&15, lane-half selects K pair {0,1}/{2,3}
//   C/D    : col n = lane&15, VGPR i holds row i + 8*(lane>>4)
// ---------------------------------------------------------------------------
__global__ void __launch_bounds__(256)
gemm_wmma_f32(const float* __restrict__ mh,     // [128 x 152]
              const float* __restrict__ W,      // [1024 x 150] row-major
              const float* __restrict__ bias,   // [1024]
              float*       __restrict__ out) {  // [128*512 x 1024]
    const int lane  = threadIdx.x & 31;
    const int wave  = threadIdx.x >> 5;
    const int tile  = blockIdx.x * 8 + wave;    // 0..511
    const int tileM = tile >> 6;                // 0..7   (image tiles)
    const int tileN = tile & 63;                // 0..63  (path-dim tiles)

    const int mrow = tileM * 16 + (lane & 15);  // image index for A loads
    const int ncol = tileN * 16 + (lane & 15);  // output dim for B loads / D
    const int kb   = (lane >> 4) * 2;           // K sub-offset per lane half

    v8f acc = {};
    #pragma unroll
    for (int k0 = 0; k0 < C_PAD; k0 += 4) {
        const int ka = k0 + kb;
        v2f a, b;
        a.x = mh[mrow * C_PAD + ka];            // MH padded with zeros at k>=150
        a.y = mh[mrow * C_PAD + ka + 1];
        // Wt[k][n] = W[n*150 + k]; guard the two pad columns (MH is 0 there).
        b.x = (ka     < C_DIM) ? W[ncol * C_DIM + ka]     : 0.0f;
        b.y = (ka + 1 < C_DIM) ? W[ncol * C_DIM + ka + 1] : 0.0f;
        acc = __builtin_amdgcn_wmma_f32_16x16x4_f32(
            /*neg_a=*/false, a, /*neg_b=*/false, b,
            /*c_mod=*/(short)0, acc, /*reuse_a=*/false, /*reuse_b=*/false);
    }

    const float bv    = bias[ncol];
    const int   mbase = tileM * 16 + (lane >> 4) * 8;
    #pragma unroll
    for (int i = 0; i < 8; ++i) {
        const int img = mbase + i;              // D row (image index)
        // write Y[img, ncol] into the p==0 row of that image
        out[(size_t)img * PAIRS * D_DIM + ncol] = acc[i] + bv;
    }
}

// ---------------------------------------------------------------------------
// Kernel 3: broadcast each image's p==0 row to rows p=1..511.
// One block per output row; 256 lanes x one float4 = 4 KB/row, fully
// coalesced. Non-temporal stores: the 256 MB stream must not thrash L2,
// while the 512 KB of source rows stay RT-cached.
// ---------------------------------------------------------------------------
__global__ void __launch_bounds__(256)
broadcast_rows(float* __restrict__ out) {
    const int row = blockIdx.x;                 // 0..65535
    const int p   = row & (PAIRS - 1);
    if (p == 0) return;                         // source row, already final
    const int img = row >> 9;                   // row / 512
    const v4f* __restrict__ src =
        (const v4f*)(out + (size_t)img * PAIRS * D_DIM);
    v4f v = src[threadIdx.x];
    v4f* dst = (v4f*)(out + (size_t)row * D_DIM);
    __builtin_nontemporal_store(v, &dst[threadIdx.x]);
}

// ---------------------------------------------------------------------------
extern "C" void kernel_launch(void* const* d_in, const int* in_sizes, int n_in,
                              void* d_out, int out_size, void* d_ws, size_t ws_size,
                              hipStream_t stream) {
    const int*   preds = (const int*)  d_in[0];   // [128*512] int32
    const float* W     = (const float*)d_in[1];   // [1024*150] fp32
    const float* bias  = (const float*)d_in[2];   // [1024] fp32
    float*       out   = (float*)d_out;           // [128*512*1024] fp32
    float*       mh    = (float*)d_ws;            // 128*152*4 = 77824 B scratch

    build_multihot<<<NUM_IMG, PAIRS, 0, stream>>>(preds, mh);
    gemm_wmma_f32 <<<64, 256, 0, stream>>>(mh, W, bias, out);
    broadcast_rows<<<NUM_IMG * PAIRS, 256, 0, stream>>>(out);
}